// BilinearInterpolation_26542897889779
// MI455X (gfx1250) — compile-verified
//
#include <hip/hip_runtime.h>

typedef float v2f __attribute__((ext_vector_type(2)));
typedef float v8f __attribute__((ext_vector_type(8)));

#define STN_H 512
#define STN_W 512

__device__ __forceinline__ int iclamp(int v, int lo, int hi) {
    return v < lo ? lo : (v > hi ? hi : v);
}

// One wave32 per 16x16 output tile. Coordinates for the whole tile are
// produced by two V_WMMA_F32_16X16X4_F32 rank-2 products (the reference's
// grid-generator einsum); each lane then bilinearly samples its 8 pixels.
__global__ __launch_bounds__(256) void stn_bilinear_kernel(
    const float* __restrict__ img,    // [B, 512, 512, 3]
    const float* __restrict__ theta,  // [B, 2, 3]
    float* __restrict__ out)          // [B, 512, 512, 3]
{
    const int lane = threadIdx.x & 31;
    const int wave = threadIdx.x >> 5;

    // 1024 tiles per batch (32x32), 8 tiles (waves) per block -> 128 blocks/batch.
    const int b           = blockIdx.x >> 7;                   // uniform per block
    const int tileInBatch = ((blockIdx.x & 127) << 3) | wave;  // 0..1023
    const int rowBase     = (tileInBatch >> 5) << 4;
    const int colBase     = (tileInBatch & 31) << 4;

    // theta is block-uniform -> scalar loads
    const float t0 = theta[b * 6 + 0];
    const float t1 = theta[b * 6 + 1];
    const float t2 = theta[b * 6 + 2];
    const float t3 = theta[b * 6 + 3];
    const float t4 = theta[b * 6 + 4];
    const float t5 = theta[b * 6 + 5];

    const int  mn = lane & 15;
    const bool hi = lane >= 16;

    // Normalized coords: linspace(-1,1,512) -> -1 + 2*i/511
    const float ysm = -1.0f + (2.0f / 511.0f) * (float)(rowBase + mn);
    const float xsn = -1.0f + (2.0f / 511.0f) * (float)(colBase + mn);

    // A (16x4 f32): lanes 0-15 hold K=0 (v0) and K=1 (v1); lanes 16-31 hold K=2,3 (zeroed).
    // B (4x16 f32): lanes 0-15 hold rows K=0 (v0), K=1 (v1); lanes 16-31 hold K=2,3 (zeroed).
    // D[m][n] = ys(m)*t1 + 1*(t0*xs(n)+t2)   (and likewise for y with t4,t3,t5)
    v2f A, Bx, By;
    A[0]  = hi ? 0.0f : ysm;
    A[1]  = hi ? 0.0f : 1.0f;
    Bx[0] = hi ? 0.0f : t1;
    Bx[1] = hi ? 0.0f : (t0 * xsn + t2);
    By[0] = hi ? 0.0f : t4;
    By[1] = hi ? 0.0f : (t3 * xsn + t5);

    v8f zero = {};
    v8f Dx = __builtin_amdgcn_wmma_f32_16x16x4_f32(false, A, false, Bx,
                                                   (short)0, zero, false, false);
    v8f Dy = __builtin_amdgcn_wmma_f32_16x16x4_f32(false, A, false, By,
                                                   (short)0, zero, false, false);

    const int    col    = colBase + mn;       // D layout: N = lane & 15
    const size_t baseB  = (size_t)b * (size_t)(STN_H * STN_W * 3);
    const float* imgB   = img + baseB;
    float*       outB   = out + baseB;

#pragma unroll
    for (int v = 0; v < 8; ++v) {
        const int row = rowBase + v + (hi ? 8 : 0);   // D layout: M = v + 8*(lane>=16)

        // Map to pixel space, exactly as the reference.
        float x = (Dx[v] + 1.0f) * (0.5f * (float)STN_W);
        float y = (Dy[v] + 1.0f) * (0.5f * (float)STN_H);

        float xf = floorf(x);
        float yf = floorf(y);
        int x0 = iclamp((int)xf,     0, STN_W - 1);
        int x1 = iclamp((int)xf + 1, 0, STN_W - 1);
        int y0 = iclamp((int)yf,     0, STN_H - 1);
        int y1 = iclamp((int)yf + 1, 0, STN_H - 1);

        float xc = fminf(fmaxf(x, 0.0f), (float)(STN_W - 1));
        float yc = fminf(fmaxf(y, 0.0f), (float)(STN_H - 1));

        float wx1 = (float)x1 - xc;   // weight toward x0 column
        float wx0 = xc - (float)x0;   // weight toward x1 column
        float wy1 = (float)y1 - yc;   // weight toward y0 row
        float wy0 = yc - (float)y0;   // weight toward y1 row

        float wa = wx1 * wy1;   // (y0, x0)
        float wb = wx1 * wy0;   // (y1, x0)
        float wc = wx0 * wy1;   // (y0, x1)
        float wd = wx0 * wy0;   // (y1, x1)

        const float* pa = imgB + ((size_t)y0 * STN_W + x0) * 3;
        const float* pb = imgB + ((size_t)y1 * STN_W + x0) * 3;
        const float* pc = imgB + ((size_t)y0 * STN_W + x1) * 3;
        const float* pd = imgB + ((size_t)y1 * STN_W + x1) * 3;
        float*       po = outB + ((size_t)row * STN_W + col) * 3;

#pragma unroll
        for (int c = 0; c < 3; ++c) {
            po[c] = wa * pa[c] + wb * pb[c] + wc * pc[c] + wd * pd[c];
        }
    }
}

extern "C" void kernel_launch(void* const* d_in, const int* in_sizes, int n_in,
                              void* d_out, int out_size, void* d_ws, size_t ws_size,
                              hipStream_t stream) {
    const float* images = (const float*)d_in[0];  // [B,512,512,3] f32
    const float* theta  = (const float*)d_in[1];  // [B,2,3] f32
    float*       out    = (float*)d_out;          // [B,512,512,3] f32

    const int B = in_sizes[1] / 6;                // 64 for the reference
    // 1024 tiles per batch, 8 waves (tiles) per 256-thread block.
    dim3 grid(B * 128);
    dim3 block(256);
    stn_bilinear_kernel<<<grid, block, 0, stream>>>(images, theta, out);
}